// FCOSPostProcessor_76321568850622
// MI455X (gfx1250) — compile-verified
//
#include <hip/hip_runtime.h>
#include <cstdint>

// FCOS post-processor, fused single-kernel design for MI455X (gfx1250).
// One workgroup (1024 threads = 32 wave32) per image; everything lives in the
// 320KB WGP LDS: staged logits (async copy), radix-select + bitonic top-1000,
// box decode, WMMA-assisted pairwise-IoU suppression matrix, bitmask NMS scan.

#define HW      16800      // H*W = 100*168
#define NIMG    32
#define K_PRE   1000
#define K_POST  100

typedef __attribute__((ext_vector_type(2))) float v2f;
typedef __attribute__((ext_vector_type(8))) float v8f;

// ---- dynamic-LDS layout (bytes). Scores alias the suppression matrix. ----
#define OFF_SCORES 0          // 16800 * 4 = 67200 B   (phase 1-2 only)
#define OFF_SUP    0          // 1024 rows * 32 u32 = 131072 B (phase 5-6)
#define OFF_HIST   131072     // 256 * 4
#define OFF_CAND   132096     // 1024 * 8 (u64, 8B aligned)
#define OFF_BX1    140288     // 1024 * 4 each
#define OFF_BY1    144384
#define OFF_BX2    148480
#define OFF_BY2    152576
#define OFF_AREA   156672
#define OFF_SC     160768
#define OFF_KEEP   164864     // 32 * 4
#define OFF_PFX    164992     // 33 * 4
#define OFF_MISC   165124     // 4 * 4
#define SMEM_BYTES 165184

__device__ __forceinline__ uint32_t fkey(float f) {
  uint32_t u = __float_as_uint(f);
  return (u & 0x80000000u) ? ~u : (u | 0x80000000u);   // ascending uint order
}
__device__ __forceinline__ float unfkey(uint32_t k) {
  uint32_t u = (k & 0x80000000u) ? (k & 0x7FFFFFFFu) : ~k;
  return __uint_as_float(u);
}

extern "C" __global__ void __launch_bounds__(1024)
fcos_post_kernel(const float* __restrict__ loc,
                 const float* __restrict__ cls,
                 const float* __restrict__ reg,
                 const int*   __restrict__ imh,
                 const int*   __restrict__ imw,
                 float*       __restrict__ out)
{
  extern __shared__ char smem[];
  float*              sScore = (float*)(smem + OFF_SCORES);
  uint32_t*           sSup   = (uint32_t*)(smem + OFF_SUP);
  uint32_t*           sHist  = (uint32_t*)(smem + OFF_HIST);
  unsigned long long* sCand  = (unsigned long long*)(smem + OFF_CAND);
  float*              sX1    = (float*)(smem + OFF_BX1);
  float*              sY1    = (float*)(smem + OFF_BY1);
  float*              sX2    = (float*)(smem + OFF_BX2);
  float*              sY2    = (float*)(smem + OFF_BY2);
  float*              sArea  = (float*)(smem + OFF_AREA);
  float*              sSc    = (float*)(smem + OFF_SC);
  uint32_t*           sKeep  = (uint32_t*)(smem + OFF_KEEP);
  uint32_t*           sPfx   = (uint32_t*)(smem + OFF_PFX);
  uint32_t*           sMisc  = (uint32_t*)(smem + OFF_MISC);

  const int tid  = threadIdx.x;
  const int lane = tid & 31;
  const int wave = tid >> 5;
  const int n    = blockIdx.x;

  // ---------- Phase 0: async-stage per-image logits global -> LDS ----------
  // No static LDS in this kernel, so the dynamic segment starts at LDS offset 0.
  {
    uint64_t base = (uint64_t)(cls + (size_t)n * HW);
    for (int e = tid; e < HW; e += 1024) {
      uint32_t ldsA = (uint32_t)(OFF_SCORES) + (uint32_t)e * 4u;
      uint32_t gOff = (uint32_t)e * 4u;
      asm volatile("global_load_async_to_lds_b32 %0, %1, %2"
                   :: "v"(ldsA), "v"(gOff), "s"(base) : "memory");
    }
    asm volatile("s_wait_asynccnt 0x0" ::: "memory");
  }
  __syncthreads();

  // ---------- Phase 1: radix-select the K_PRE-th largest key (threshold T) ----------
  uint32_t prefix = 0, kneed = K_PRE;
  for (int shift = 24; shift >= 0; shift -= 8) {
    for (int b = tid; b < 256; b += 1024) sHist[b] = 0;
    __syncthreads();
    uint32_t hiMask = (shift == 24) ? 0u : (0xFFFFFFFFu << (shift + 8));
    for (int e = tid; e < HW; e += 1024) {
      uint32_t k = fkey(sScore[e]);
      if ((k & hiMask) == prefix) atomicAdd(&sHist[(k >> shift) & 255u], 1u);
    }
    __syncthreads();
    if (tid == 0) {
      uint32_t c = 0; int b = 255;
      for (; b > 0; --b) { uint32_t h = sHist[b]; if (c + h >= kneed) break; c += h; }
      sMisc[0] = (uint32_t)b; sMisc[1] = c;
    }
    __syncthreads();
    prefix |= sMisc[0] << shift;
    kneed  -= sMisc[1];
    __syncthreads();
  }
  const uint32_t T = prefix;              // exact key of the K_PRE-th largest
  const uint32_t gtTotal = K_PRE - kneed; // #keys strictly > T

  // ---------- Phase 2: compact top-K_PRE candidates into slots 0..999 ----------
  if (tid == 0) { sMisc[2] = 0; sMisc[3] = 0; }
  __syncthreads();
  for (int e = tid; e < HW; e += 1024) {
    uint32_t k = fkey(sScore[e]);
    unsigned long long pk =
        ((unsigned long long)k << 32) | (uint32_t)(~(uint32_t)e); // ~idx: ties -> smaller idx first
    if (k > T) {
      uint32_t p = atomicAdd(&sMisc[2], 1u);
      sCand[p] = pk;
    } else if (k == T) {
      uint32_t q = atomicAdd(&sMisc[3], 1u);
      if (q < kneed) sCand[gtTotal + q] = pk;
    }
  }
  if (tid >= K_PRE) sCand[tid] = 0ULL;    // sentinel padding, sinks in sort
  __syncthreads();

  // ---------- Phase 3: bitonic sort 1024 u64 keys, descending ----------
  for (unsigned k = 2; k <= 1024; k <<= 1) {
    for (unsigned j = k >> 1; j > 0; j >>= 1) {
      __syncthreads();
      unsigned i = (unsigned)tid, ixj = i ^ j;
      if (ixj > i) {
        unsigned long long a = sCand[i], b = sCand[ixj];
        bool sw = ((i & k) == 0) ? (a < b) : (a > b);
        if (sw) { sCand[i] = b; sCand[ixj] = a; }
      }
    }
  }
  __syncthreads();

  // ---------- Phase 4: decode boxes, validity, init keep mask; zero sup-matrix ----------
  const float imgW = (float)imw[0];
  const float imgH = (float)imh[0];
  {
    unsigned long long kv = sCand[tid];
    uint32_t k32 = (uint32_t)(kv >> 32);
    uint32_t e   = ~(uint32_t)kv;
    bool real = (tid < K_PRE) && (kv != 0ULL);
    if (!real) e = 0;
    float lg = unfkey(k32);
    float sc = __frcp_rn(1.0f + __expf(-lg));     // sigmoid, fast path
    float lx = loc[2 * e], ly = loc[2 * e + 1];
    float r0 = reg[((size_t)n * 4 + 0) * HW + e];
    float r1 = reg[((size_t)n * 4 + 1) * HW + e];
    float r2 = reg[((size_t)n * 4 + 2) * HW + e];
    float r3 = reg[((size_t)n * 4 + 3) * HW + e];
    float x1 = fminf(fmaxf(lx - r0, 0.f), imgW - 1.f);
    float y1 = fminf(fmaxf(ly - r1, 0.f), imgH - 1.f);
    float x2 = fminf(fmaxf(lx + r2, 0.f), imgW - 1.f);
    float y2 = fminf(fmaxf(ly + r3, 0.f), imgH - 1.f);
    bool valid = real && (sc > 0.01f) &&
                 (x2 - x1 + 1.0f >= 0.0f) && (y2 - y1 + 1.0f >= 0.0f);
    sX1[tid] = x1; sY1[tid] = y1; sX2[tid] = x2; sY2[tid] = y2;
    sArea[tid] = fmaxf(x2 - x1, 0.f) * fmaxf(y2 - y1, 0.f);
    sSc[tid] = sc;
    uint32_t m = __builtin_amdgcn_ballot_w32(valid);
    if (lane == 0) sKeep[wave] = m;
  }
  for (int w = tid; w < 1024 * 32; w += 1024) sSup[w] = 0; // scores are dead now
  __syncthreads();

  // ---------- Phase 5: suppression matrix in 16x16 tiles (upper triangle) ----------
  // union base term area_i + area_j is a rank-2 outer sum -> V_WMMA_F32_16X16X4_F32:
  //   A[M][:] = [area_row[M], 1, 0, 0],  B[:][N] = [1, area_col[N], 0, 0]^T
  for (int tile = wave; tile < 64 * 64; tile += 32) {   // wave-uniform: EXEC all-ones at WMMA
    int tr = tile >> 6, tc = tile & 63;
    if (tc < tr) continue;
    int rb = tr << 4, cb = tc << 4;
    int  l15 = lane & 15;
    bool lo  = lane < 16;
    v2f a, b;
    a.x = lo ? sArea[rb + l15] : 0.0f;   // K=0 (lanes 0-15) / K=2 (lanes 16-31)
    a.y = lo ? 1.0f            : 0.0f;   // K=1 / K=3
    b.x = lo ? 1.0f            : 0.0f;
    b.y = lo ? sArea[cb + l15] : 0.0f;
    v8f c = {0.f,0.f,0.f,0.f,0.f,0.f,0.f,0.f};
    v8f d = __builtin_amdgcn_wmma_f32_16x16x4_f32(false, a, false, b, (short)0, c,
                                                  false, false);
    int   nj  = cb + l15;                // this lane's column
    float cx1 = sX1[nj], cy1 = sY1[nj], cx2 = sX2[nj], cy2 = sY2[nj];
#pragma unroll
    for (int r = 0; r < 8; ++r) {
      int   mi  = rb + r + (lo ? 0 : 8); // D vgpr r holds rows M=r (lanes<16) / M=r+8
      float rx1 = sX1[mi], ry1 = sY1[mi], rx2 = sX2[mi], ry2 = sY2[mi];
      float iw = fmaxf(fminf(rx2, cx2) - fmaxf(rx1, cx1), 0.f);
      float ih = fmaxf(fminf(ry2, cy2) - fmaxf(ry1, cy1), 0.f);
      float inter = iw * ih;
      float uni   = d[r] - inter;        // area_i + area_j - inter (WMMA result)
      // iou > 0.5  <=>  2*inter > max(uni, 1e-9): divide-free comparison
      bool  sup   = ((inter + inter) > fmaxf(uni, 1e-9f)) && (nj > mi);
      uint32_t msk = __builtin_amdgcn_ballot_w32(sup);
      if (lane == 0) {
        uint32_t lo16 = msk & 0xFFFFu;   // row rb+r,   cols cb..cb+15
        uint32_t hi16 = msk >> 16;       // row rb+8+r, cols cb..cb+15
        int wsel = tc >> 1;
        int sh   = (tc & 1) << 4;
        if (lo16) atomicOr(&sSup[(rb + r)     * 32 + wsel], lo16 << sh);
        if (hi16) atomicOr(&sSup[(rb + 8 + r) * 32 + wsel], hi16 << sh);
      }
    }
  }
  __syncthreads();

  // ---------- Phase 6: sequential greedy keep-scan (one wave, registers) ----------
  if (tid < 32) {
    uint32_t k = sKeep[lane];
    for (int i = 0; i < K_PRE; ++i) {
      uint32_t wk = (uint32_t)__builtin_amdgcn_readlane((int)k, i >> 5);
      if ((wk >> (i & 31)) & 1u) k &= ~sSup[i * 32 + lane];
    }
    sKeep[lane] = k;
  }
  __syncthreads();

  // ---------- Phase 7: stable placement of first K_POST rows ----------
  if (tid == 0) {
    uint32_t a = 0;
    for (int w = 0; w < 32; ++w) { sPfx[w] = a; a += __popc(sKeep[w]); }
    sPfx[32] = a;
  }
  __syncthreads();
  if (tid < K_PRE) {
    int w = tid >> 5, bpos = tid & 31;
    uint32_t word = sKeep[w];
    bool     kept = (word >> bpos) & 1u;
    uint32_t before  = sPfx[w] + __popc(word & ((1u << bpos) - 1u));
    uint32_t numKept = sPfx[32];
    uint32_t pos = kept ? before : (numKept + ((uint32_t)tid - before));
    if (pos < K_POST) {
      float* o = out + ((size_t)n * K_POST + pos) * 6;
      o[0] = sX1[tid]; o[1] = sY1[tid]; o[2] = sX2[tid]; o[3] = sY2[tid];
      o[4] = kept ? sSc[tid] : 0.0f;
      o[5] = 1.0f;                       // C==1 -> all labels are 1.0
    }
  }
}

extern "C" void kernel_launch(void* const* d_in, const int* in_sizes, int n_in,
                              void* d_out, int out_size, void* d_ws, size_t ws_size,
                              hipStream_t stream) {
  (void)in_sizes; (void)n_in; (void)out_size; (void)d_ws; (void)ws_size;
  const float* loc = (const float*)d_in[0];
  const float* cls = (const float*)d_in[1];
  const float* reg = (const float*)d_in[2];
  // d_in[3] (centerness) is computed-and-discarded by the reference.
  const int*   imh = (const int*)d_in[4];
  const int*   imw = (const int*)d_in[5];

  (void)hipFuncSetAttribute((const void*)fcos_post_kernel,
                            hipFuncAttributeMaxDynamicSharedMemorySize, SMEM_BYTES);
  fcos_post_kernel<<<dim3(NIMG), dim3(1024), SMEM_BYTES, stream>>>(
      loc, cls, reg, imh, imw, (float*)d_out);
}